// IMMJumpNet_55929064129110
// MI455X (gfx1250) — compile-verified
//
#include <hip/hip_runtime.h>
#include <cmath>

#define B_N   8192
#define INDIM 64
#define HID   256
#define PI_F  3.14159265358979323846f

typedef __attribute__((ext_vector_type(2))) float v2f;
typedef __attribute__((ext_vector_type(4))) float v4f;
typedef __attribute__((ext_vector_type(8))) float v8f;

// ---------------- prep: per-row scalars ----------------
__global__ void prep_kernel(const float* __restrict__ y, const float* __restrict__ s_rand,
                            const float* __restrict__ eps, const float* __restrict__ y_tmp_init,
                            const float* __restrict__ noise_t,
                            float* __restrict__ cosv, float* __restrict__ sinv,
                            float* __restrict__ ysr, float* __restrict__ ytt,
                            float* __restrict__ yA, float* __restrict__ partials)
{
    if (blockIdx.x == 0 && threadIdx.x < 3) partials[threadIdx.x] = 0.f;
    int m = blockIdx.x * blockDim.x + threadIdx.x;
    if (m >= B_N) return;
    float s  = s_rand[m] * 0.9f + 0.1f;          // s_rand*(1-STEP)+STEP
    float yv = y[m];
    ysr[m]  = sqrtf(1.0f - s) * yv + sqrtf(s) * eps[m];   // SIGMA=1
    ytt[m]  = yv + 0.001f * noise_t[m];
    yA[m]   = y_tmp_init[m];                     // * SIGMA(=1)
    cosv[m] = cosf(PI_F * s);
    sinv[m] = sinf(PI_F * s);
}

// ------- precompute rank-2 time-embedding fold into layer1 bias -------
__global__ void precomp_kernel(const float* __restrict__ Wt, const float* __restrict__ bt,
                               const float* __restrict__ W1, const float* __restrict__ b1,
                               float* __restrict__ c0, float* __restrict__ c1, float* __restrict__ c2)
{
    int j = threadIdx.x;                         // 256 threads
    float a0 = 0.f, a1 = 0.f, ab = 0.f;
    for (int k = 0; k < 64; ++k) {
        float w = W1[(65 + k) * HID + j];
        a0 += Wt[k]      * w;
        a1 += Wt[64 + k] * w;
        ab += bt[k]      * w;
    }
    c1[j] = a0; c2[j] = a1; c0[j] = ab + b1[j];
}

// ------- repack weights into WMMA B-fragment lane order -------
// Wp[ct][ktp][lane] = {W[kA,n], W[kA+1,n], W[kB,n], W[kB+1,n]}
//   n = ct*16 + (lane&15), kk = (lane>>4)*2, kA = (2*ktp)*4+kk, kB = (2*ktp+1)*4+kk
__global__ void pack_kernel(const float* __restrict__ W, int nktp, v4f* __restrict__ out)
{
    int tid = blockIdx.x * blockDim.x + threadIdx.x;
    if (tid >= 16 * nktp * 32) return;
    int lane = tid & 31;
    int ktp  = (tid >> 5) % nktp;
    int ct   = tid / (nktp * 32);
    int kk   = (lane >> 4) * 2;
    int n    = ct * 16 + (lane & 15);
    int kA   = (ktp * 2)     * 4 + kk;
    int kB   = (ktp * 2 + 1) * 4 + kk;
    v4f v;
    v[0] = W[(kA    ) * HID + n];
    v[1] = W[(kA + 1) * HID + n];
    v[2] = W[(kB    ) * HID + n];
    v[3] = W[(kB + 1) * HID + n];
    out[tid] = v;
}

// ---------------- fused MLP forward (layers 1,2,3) ----------------
#define MT     32          // rows per block (2 waves x 16)
#define SX_LD  68          // padded LDS strides (bank-conflict avoidance)
#define SH_LD  260

__global__ __launch_bounds__(64) void mlp_kernel(
    const float* __restrict__ X,  const float* __restrict__ yin,
    const float* __restrict__ cosv, const float* __restrict__ sinv,
    float uc, float us, int per_row,
    const v4f* __restrict__ Wp1, const v4f* __restrict__ Wp2,
    const float* __restrict__ W1, const float* __restrict__ c0,
    const float* __restrict__ c1, const float* __restrict__ c2,
    const float* __restrict__ b2,
    const float* __restrict__ W3, const float* __restrict__ b3,
    float* __restrict__ yout)
{
    __shared__ float sX[MT * SX_LD];
    __shared__ float sH[MT * SH_LD];
    __shared__ float sC[MT], sS[MT], sY[MT];

    const int tid  = threadIdx.x;
    const int lane = tid & 31;
    const int wave = tid >> 5;
    const int rowBase = blockIdx.x * MT;

    // stage X tile (coalesced)
    for (int idx = tid; idx < MT * INDIM; idx += 64) {
        int r = idx >> 6, c = idx & 63;
        sX[r * SX_LD + c] = X[(rowBase + r) * INDIM + c];
    }
    if (tid < MT) {
        int gm = rowBase + tid;
        sY[tid] = yin[gm];
        sC[tid] = per_row ? cosv[gm] : uc;
        sS[tid] = per_row ? sinv[gm] : us;
    }
    __syncthreads();

    const int r16 = lane & 15;
    const int kk  = (lane >> 4) * 2;      // A-frag K sub-offset per ISA layout
    const int lr  = wave * 16 + r16;      // local A row

    // ---- layer 1: h1 = relu(X@W1[0:64] + y*W1[64] + cos*c1 + sin*c2 + c0) ----
    for (int ct = 0; ct < 16; ++ct) {
        const v4f* wp = Wp1 + (ct * 8) * 32 + lane;
        v8f acc = {0.f,0.f,0.f,0.f,0.f,0.f,0.f,0.f};
        #pragma unroll
        for (int ktp = 0; ktp < 8; ++ktp) {
            v4f bv = wp[ktp * 32];                     // coalesced b128
            const int k0 = ktp * 8 + kk;
            v2f a0, a1, b0, b1;
            a0[0] = sX[lr * SX_LD + k0];     a0[1] = sX[lr * SX_LD + k0 + 1];
            a1[0] = sX[lr * SX_LD + k0 + 4]; a1[1] = sX[lr * SX_LD + k0 + 5];
            b0[0] = bv[0]; b0[1] = bv[1];
            b1[0] = bv[2]; b1[1] = bv[3];
            acc = __builtin_amdgcn_wmma_f32_16x16x4_f32(false, a0, false, b0,
                                                        (short)0, acc, false, false);
            acc = __builtin_amdgcn_wmma_f32_16x16x4_f32(false, a1, false, b1,
                                                        (short)0, acc, false, false);
        }
        const int   n   = ct * 16 + r16;
        const float w1y = W1[64 * HID + n];
        const float b0v = c0[n], b1v = c1[n], b2v = c2[n];
        #pragma unroll
        for (int v = 0; v < 8; ++v) {
            const int ml = wave * 16 + ((lane >> 4) ? (v + 8) : v);
            float val = acc[v] + b0v + sC[ml] * b1v + sS[ml] * b2v + sY[ml] * w1y;
            sH[ml * SH_LD + n] = val > 0.f ? val : 0.f;
        }
    }
    __syncthreads();

    // ---- layer 2 + layer 3 fused: out = relu(h1@W2+b2) @ W3 + b3 ----
    float pr[8] = {0.f,0.f,0.f,0.f,0.f,0.f,0.f,0.f};
    const float b3v = b3[0];
    for (int ct = 0; ct < 16; ++ct) {
        const v4f* wp = Wp2 + (ct * 32) * 32 + lane;
        v8f acc = {0.f,0.f,0.f,0.f,0.f,0.f,0.f,0.f};
        #pragma unroll 8
        for (int ktp = 0; ktp < 32; ++ktp) {
            v4f bv = wp[ktp * 32];                     // coalesced b128
            const int k0 = ktp * 8 + kk;
            v2f a0, a1, b0, b1;
            a0[0] = sH[lr * SH_LD + k0];     a0[1] = sH[lr * SH_LD + k0 + 1];
            a1[0] = sH[lr * SH_LD + k0 + 4]; a1[1] = sH[lr * SH_LD + k0 + 5];
            b0[0] = bv[0]; b0[1] = bv[1];
            b1[0] = bv[2]; b1[1] = bv[3];
            acc = __builtin_amdgcn_wmma_f32_16x16x4_f32(false, a0, false, b0,
                                                        (short)0, acc, false, false);
            acc = __builtin_amdgcn_wmma_f32_16x16x4_f32(false, a1, false, b1,
                                                        (short)0, acc, false, false);
        }
        const int   n  = ct * 16 + r16;
        const float bb = b2[n];
        const float w3 = W3[n];
        #pragma unroll
        for (int v = 0; v < 8; ++v) {
            float h = acc[v] + bb;
            h = h > 0.f ? h : 0.f;
            pr[v] += h * w3;
        }
    }
    // cross-lane reduce over the 16 lanes sharing each row
    #pragma unroll
    for (int v = 0; v < 8; ++v) {
        float p = pr[v];
        p += __shfl_xor(p, 1, 32);
        p += __shfl_xor(p, 2, 32);
        p += __shfl_xor(p, 4, 32);
        p += __shfl_xor(p, 8, 32);
        if (r16 == 0) {
            int m = rowBase + wave * 16 + (lane >> 4) * 8 + v;
            yout[m] = p + b3v;
        }
    }
}

// ---------------- select y_s_real = where(use_model, y_tmp, y_s_real) ----------------
__global__ void select_kernel(const float* __restrict__ yA, const float* __restrict__ ysr,
                              const int* __restrict__ use_model, float* __restrict__ yFin)
{
    int m = blockIdx.x * blockDim.x + threadIdx.x;
    if (m < B_N) yFin[m] = use_model[m] ? yA[m] : ysr[m];
}

// ---------------- Laplace MMD^2 pairwise sums ----------------
__global__ __launch_bounds__(128) void mmd_kernel(const float* __restrict__ yh,
                                                  const float* __restrict__ yt,
                                                  float* __restrict__ partials)
{
    __shared__ float tH[256], tT[256];
    const int tid = threadIdx.x;
    const int i   = blockIdx.x * 128 + tid;
    const float xi = yh[i], ti = yt[i];
    float sxx = 0.f, syy = 0.f, sxy = 0.f;
    for (int j0 = 0; j0 < B_N; j0 += 256) {
        __syncthreads();
        for (int k = tid; k < 256; k += 128) { tH[k] = yh[j0 + k]; tT[k] = yt[j0 + k]; }
        __syncthreads();
        #pragma unroll 4
        for (int k = 0; k < 256; ++k) {
            sxx += __expf(-fabsf(xi - tH[k]));   // gamma = 1
            syy += __expf(-fabsf(ti - tT[k]));
            sxy += __expf(-fabsf(xi - tT[k]));
        }
    }
    atomicAdd(&partials[0], sxx);
    atomicAdd(&partials[1], syy);
    atomicAdd(&partials[2], sxy);
}

__global__ void finalize_kernel(const float* __restrict__ partials, float* __restrict__ out)
{
    const float m = (float)B_N;
    float txx = (partials[0] - m) / (m * (m - 1.f));
    float tyy = (partials[1] - m) / (m * (m - 1.f));
    out[0] = txx + tyy - 2.f * partials[2] / (m * m);
}

extern "C" void kernel_launch(void* const* d_in, const int* in_sizes, int n_in,
                              void* d_out, int out_size, void* d_ws, size_t ws_size,
                              hipStream_t stream)
{
    const float* x          = (const float*)d_in[0];
    const float* y          = (const float*)d_in[1];
    const float* s_rand     = (const float*)d_in[2];
    const float* eps        = (const float*)d_in[3];
    const int*   use_model  = (const int*)  d_in[4];
    const float* y_tmp_init = (const float*)d_in[5];
    const float* noise_t    = (const float*)d_in[6];
    const float* Wt         = (const float*)d_in[7];
    const float* bt         = (const float*)d_in[8];
    const float* W1         = (const float*)d_in[9];
    const float* b1         = (const float*)d_in[10];
    const float* W2         = (const float*)d_in[11];
    const float* b2         = (const float*)d_in[12];
    const float* W3         = (const float*)d_in[13];
    const float* b3         = (const float*)d_in[14];

    float* ws   = (float*)d_ws;
    float* cosv = ws;            float* sinv = ws + 8192;
    float* ysr  = ws + 16384;    float* ytt  = ws + 24576;
    float* yA   = ws + 32768;    float* yB   = ws + 40960;
    float* yFin = ws + 49152;    float* yHat = ws + 57344;
    float* c0   = ws + 65536;    float* c1v  = c0 + 256;
    float* c2v  = c1v + 256;     float* partials = c2v + 256;   // 3 floats @ 66304
    v4f*   Wp1  = (v4f*)(ws + 66320);                 // 16*8*32  v4f = 64 KB
    v4f*   Wp2  = (v4f*)(ws + 66320 + 16384);         // 16*32*32 v4f = 256 KB

    prep_kernel<<<B_N / 256, 256, 0, stream>>>(y, s_rand, eps, y_tmp_init, noise_t,
                                               cosv, sinv, ysr, ytt, yA, partials);
    precomp_kernel<<<1, 256, 0, stream>>>(Wt, bt, W1, b1, c0, c1v, c2v);
    pack_kernel<<<(16 * 8  * 32 + 255) / 256, 256, 0, stream>>>(W1, 8,  Wp1);
    pack_kernel<<<(16 * 32 * 32 + 255) / 256, 256, 0, stream>>>(W2, 32, Wp2);

    float* bufs[2] = { yA, yB };
    for (int it = 0; it < 10; ++it) {
        float sc = 1.0f - 0.1f * (float)it;     // curr_s: 1.0, 0.9, ..., 0.1 (uniform)
        float uc = cosf(PI_F * sc);
        float us = sinf(PI_F * sc);
        mlp_kernel<<<B_N / MT, 64, 0, stream>>>(x, bufs[it & 1], cosv, sinv, uc, us, 0,
                                                Wp1, Wp2, W1, c0, c1v, c2v, b2, W3, b3,
                                                bufs[(it + 1) & 1]);
    }
    // after 10 iterations result is back in yA
    select_kernel<<<B_N / 256, 256, 0, stream>>>(yA, ysr, use_model, yFin);
    // final pass with per-row s
    mlp_kernel<<<B_N / MT, 64, 0, stream>>>(x, yFin, cosv, sinv, 0.f, 0.f, 1,
                                            Wp1, Wp2, W1, c0, c1v, c2v, b2, W3, b3, yHat);
    mmd_kernel<<<B_N / 128, 128, 0, stream>>>(yHat, ytt, partials);
    finalize_kernel<<<1, 1, 0, stream>>>(partials, (float*)d_out);
}